// DIoULoss_15006615733671
// MI455X (gfx1250) — compile-verified
//
#include <hip/hip_runtime.h>

typedef __attribute__((ext_vector_type(2))) float v2f;
typedef __attribute__((ext_vector_type(4))) float v4f;
typedef __attribute__((ext_vector_type(8))) float v8f;

#define DIOU_EPS 1e-6f
#define BLOCKS   2048
#define THREADS  256
#define WAVES    (THREADS / 32)

// Full 32-lane (wave32) sum using V_WMMA_F32_16X16X4_F32.
// A[m][0] = p from lane m (0..15), A[m][2] = p from lane m+16, other K = 0.
// B = ones(4x16)  =>  D[m][n] = p_m + p_{m+16} for all n.
// C/D layout: c[v] holds row M=v (lanes 0-15) / M=v+8 (lanes 16-31), so
// sum_v c[v] = (p_0..p_7 + p_16..p_23) on lanes<16 and (p_8..p_15 + p_24..p_31)
// on lanes>=16; one shfl_xor(16) yields the full wave sum in every lane.
__device__ __forceinline__ float wave_sum_wmma(float p) {
    v2f a; a[0] = p;    a[1] = 0.0f;
    v2f b; b[0] = 1.0f; b[1] = 1.0f;
    v8f c = {};
    c = __builtin_amdgcn_wmma_f32_16x16x4_f32(
        /*neg_a=*/false, a, /*neg_b=*/false, b,
        /*c_mod=*/(short)0, c, /*reuse_a=*/false, /*reuse_b=*/false);
    float t = ((c[0] + c[1]) + (c[2] + c[3])) + ((c[4] + c[5]) + (c[6] + c[7]));
    t += __shfl_xor(t, 16, 32);
    return t;
}

__device__ __forceinline__ float block_sum(float acc, float* lds) {
    float wsum = wave_sum_wmma(acc);
    const int lane = threadIdx.x & 31;
    const int wave = threadIdx.x >> 5;
    if (lane == 0) lds[wave] = wsum;
    __syncthreads();
    float s = 0.0f;
    if (threadIdx.x == 0) {
        #pragma unroll
        for (int w = 0; w < WAVES; ++w) s += lds[w];
    }
    return s; // valid on thread 0 only
}

__global__ void diou_partial_kernel(const v4f* __restrict__ pred,
                                    const v4f* __restrict__ targ,
                                    float* __restrict__ partials, int n) {
    const int tid    = blockIdx.x * blockDim.x + threadIdx.x;
    const int stride = gridDim.x * blockDim.x;
    float acc = 0.0f;
    for (int i = tid; i < n; i += stride) {
        // 2x GLOBAL_LOAD_B128, non-temporal (single-pass stream > L2)
        v4f p = __builtin_nontemporal_load(&pred[i]);
        v4f q = __builtin_nontemporal_load(&targ[i]);

        float area1 = (p.z - p.x) * (p.w - p.y);
        float area2 = (q.z - q.x) * (q.w - q.y);

        float ltx = fmaxf(p.x, q.x), lty = fmaxf(p.y, q.y);
        float rbx = fminf(p.z, q.z), rby = fminf(p.w, q.w);
        float wx = fmaxf(rbx - ltx, 0.0f);
        float wy = fmaxf(rby - lty, 0.0f);
        float inter = wx * wy;
        float uni   = area1 + area2 - inter;
        float iou   = inter / fmaxf(uni, DIOU_EPS);

        // center-distance term (centers scaled by 0.5; difference of sums works too)
        float dx = (p.x + p.z) * 0.5f - (q.x + q.z) * 0.5f;
        float dy = (p.y + p.w) * 0.5f - (q.y + q.w) * 0.5f;
        float cdist = dx * dx + dy * dy;

        // enclosing-box diagonal
        float ex = fmaxf(p.z, q.z) - fminf(p.x, q.x);
        float ey = fmaxf(p.w, q.w) - fminf(p.y, q.y);
        float diag = ex * ex + ey * ey;

        float diou = iou - cdist / fmaxf(diag, DIOU_EPS);
        acc += (1.0f - diou);
    }

    __shared__ float lds[WAVES];
    float s = block_sum(acc, lds);
    if (threadIdx.x == 0) partials[blockIdx.x] = s;
}

__global__ void diou_final_kernel(const float* __restrict__ partials,
                                  float* __restrict__ out, int nparts, float scale) {
    float acc = 0.0f;
    for (int i = threadIdx.x; i < nparts; i += blockDim.x)
        acc += partials[i];

    __shared__ float lds[WAVES];
    float s = block_sum(acc, lds);
    if (threadIdx.x == 0) out[0] = s * scale;
}

extern "C" void kernel_launch(void* const* d_in, const int* in_sizes, int n_in,
                              void* d_out, int out_size, void* d_ws, size_t ws_size,
                              hipStream_t stream) {
    const v4f* pred = (const v4f*)d_in[0];   // predictions: (N,4) f32
    const v4f* targ = (const v4f*)d_in[1];   // targets:     (N,4) f32
    const int n = in_sizes[0] / 4;           // number of boxes
    float* partials = (float*)d_ws;          // BLOCKS floats of scratch

    diou_partial_kernel<<<BLOCKS, THREADS, 0, stream>>>(pred, targ, partials, n);
    diou_final_kernel<<<1, THREADS, 0, stream>>>(partials, (float*)d_out,
                                                 BLOCKS, 1.0f / (float)n);
}